// PlaylistGAT_88691074663135
// MI455X (gfx1250) — compile-verified
//
#include <hip/hip_runtime.h>

#define NPL 50000
#define NTR 200000
#define NAR 20000
#define NGE 500

typedef float v2f __attribute__((ext_vector_type(2)));
typedef float v8f __attribute__((ext_vector_type(8)));

#if __has_builtin(__builtin_amdgcn_tensor_load_to_lds) && \
    __has_builtin(__builtin_amdgcn_s_wait_tensorcnt)
#define HAVE_TDM 1
typedef unsigned int u32x4 __attribute__((ext_vector_type(4)));
typedef int i32x4 __attribute__((ext_vector_type(4)));
typedef int i32x8 __attribute__((ext_vector_type(8)));

// DMA a [kc x 64] f32 tile (row stride 64 elems) from global into LDS via the
// Tensor Data Mover. D# layout per CDNA5 ISA.
__device__ __forceinline__ void tdm_load_w(const float* gsrc, unsigned ldsOff, int kc) {
  unsigned long long ga = (unsigned long long)(uintptr_t)gsrc;
  u32x4 g0;
  g0.x = 1u;                                              // count=1, user descriptor
  g0.y = ldsOff;                                          // lds_addr
  g0.z = (unsigned)(ga & 0xffffffffu);                    // global_addr[31:0]
  g0.w = (unsigned)((ga >> 32) & 0x01ffffffu) | (2u << 30); // addr[56:32] | type=2
  i32x8 g1;
  g1[0] = (int)(2u << 16);                                // data_size=4B
  g1[1] = (int)(64u << 16);                               // tensor_dim0=64 (lo16)
  g1[2] = (int)(((unsigned)kc & 0xffffu) << 16);          // dim0 hi | tensor_dim1 lo16
  g1[3] = (int)(64u << 16);                               // dim1 hi | tile_dim0=64
  g1[4] = (int)((unsigned)kc & 0xffffu);                  // tile_dim1=kc, tile_dim2=0
  g1[5] = 64;                                             // tensor_dim0_stride=64
  g1[6] = 0;
  g1[7] = 0;
  i32x4 z4 = {0, 0, 0, 0};
#if defined(__clang_major__) && (__clang_major__ >= 23)
  i32x8 z8 = {0, 0, 0, 0, 0, 0, 0, 0};
  __builtin_amdgcn_tensor_load_to_lds(g0, g1, z4, z4, z8, 0);
#else
  __builtin_amdgcn_tensor_load_to_lds(g0, g1, z4, z4, 0);
#endif
}
#endif

// ---------------- fill ----------------
__global__ void fill_f32(float* __restrict__ p, float v, int n) {
  int i = blockIdx.x * blockDim.x + threadIdx.x;
  if (i < n) p[i] = v;
}

// ---------------- WMMA f32 GEMM: Y[M,64] = X[M,K] @ W[K,64] (+bias) ----------------
// Block = 256 threads = 8 waves. Each wave computes a 16-row x 64-col stripe with
// 4 accumulators (A fragment reused 4x). W is staged into LDS in 64-row K-chunks
// (TDM DMA if available, else cooperative loads) and shared by all 8 waves.
__global__ __launch_bounds__(256) void wmma_gemm_f32(
    const float* __restrict__ X, const float* __restrict__ W,
    const float* __restrict__ bias, float* __restrict__ Y, int M, int K) {
  __shared__ float sW[64 * 64];
  const int lane = threadIdx.x & 31;
  const int waveInBlk = threadIdx.x >> 5;
  const int rowTiles = (M + 15) >> 4;
  const int tM = blockIdx.x * 8 + waveInBlk;
  const int half = lane >> 4;         // 0: K={k,k+1}  1: K={k+2,k+3}
  const int l15 = lane & 15;
  int arow = tM * 16 + l15;
  if (arow >= M) arow = M - 1;        // clamp: no EXEC games, junk rows masked at store
  const float* Xrow = X + (size_t)arow * K;

  v8f c0 = {0.f,0.f,0.f,0.f,0.f,0.f,0.f,0.f};
  v8f c1 = c0, c2 = c0, c3 = c0;

  for (int k0 = 0; k0 < K; k0 += 64) {
    int kc = K - k0; if (kc > 64) kc = 64;
    __syncthreads();                  // sW safe to overwrite
#ifdef HAVE_TDM
    if (waveInBlk == 0) {
      tdm_load_w(W + (size_t)k0 * 64, (unsigned)(uintptr_t)(void*)sW, kc);
      __builtin_amdgcn_s_wait_tensorcnt((short)0);
    }
#else
    for (int t = threadIdx.x; t < kc * 64; t += 256)
      sW[t] = W[(size_t)k0 * 64 + t];
#endif
    __syncthreads();                  // sW ready
    if (k0 + 64 < K) __builtin_prefetch(Xrow + k0 + 64, 0, 1);
    for (int kk = 0; kk < kc; kk += 4) {
      const int ka = kk + half * 2;
      v2f a;
      a.x = Xrow[k0 + ka];
      a.y = Xrow[k0 + ka + 1];
      const float* ws0 = &sW[ka * 64];
      const float* ws1 = &sW[(ka + 1) * 64];
      v2f b0v, b1v, b2v, b3v;
      b0v.x = ws0[l15];      b0v.y = ws1[l15];
      b1v.x = ws0[16 + l15]; b1v.y = ws1[16 + l15];
      b2v.x = ws0[32 + l15]; b2v.y = ws1[32 + l15];
      b3v.x = ws0[48 + l15]; b3v.y = ws1[48 + l15];
      c0 = __builtin_amdgcn_wmma_f32_16x16x4_f32(false, a, false, b0v, (short)0, c0, false, false);
      c1 = __builtin_amdgcn_wmma_f32_16x16x4_f32(false, a, false, b1v, (short)0, c1, false, false);
      c2 = __builtin_amdgcn_wmma_f32_16x16x4_f32(false, a, false, b2v, (short)0, c2, false, false);
      c3 = __builtin_amdgcn_wmma_f32_16x16x4_f32(false, a, false, b3v, (short)0, c3, false, false);
    }
  }

  if (tM >= rowTiles) return;
  float bb0 = 0.f, bb1 = 0.f, bb2 = 0.f, bb3 = 0.f;
  if (bias) { bb0 = bias[l15]; bb1 = bias[16 + l15]; bb2 = bias[32 + l15]; bb3 = bias[48 + l15]; }
#pragma unroll
  for (int r = 0; r < 8; ++r) {
    int orow = tM * 16 + r + half * 8;   // VGPR r: M=r (lanes 0-15) / r+8 (lanes 16-31)
    if (orow < M) {
      float* yr = Y + (size_t)orow * 64;
      yr[l15]      = c0[r] + bb0;
      yr[16 + l15] = c1[r] + bb1;
      yr[32 + l15] = c2[r] + bb2;
      yr[48 + l15] = c3[r] + bb3;
    }
  }
}

// ---------------- tiny-K encoder GEMM (K=8/7/1) ----------------
__global__ void small_gemm(const float* __restrict__ X, const float* __restrict__ W,
                           const float* __restrict__ b, float* __restrict__ Y,
                           int M, int K) {
  int i = blockIdx.x * blockDim.x + threadIdx.x;
  if (i >= M * 64) return;
  int m = i >> 6, c = i & 63;
  float acc = b ? b[c] : 0.f;
  for (int k = 0; k < K; ++k) acc += X[(size_t)m * K + k] * W[(size_t)k * 64 + c];
  Y[i] = acc;
}

// ---------------- concat [emb[idx] , feat] -> [M,128] ----------------
__global__ void concat_emb_feat(const float* __restrict__ emb, const int* __restrict__ idx,
                                const float* __restrict__ feat, float* __restrict__ out,
                                int M) {
  int i = blockIdx.x * blockDim.x + threadIdx.x;
  if (i >= M * 128) return;
  int m = i >> 7, c = i & 127;
  out[i] = (c < 64) ? emb[(size_t)idx[m] * 64 + c]
                    : feat[(size_t)m * 64 + (c - 64)];
}

// ---------------- attention logits: al[n,h] = sum_c h[n,h*16+c]*a[h,c] ----------------
__global__ void attn_logits(const float* __restrict__ hm, const float* __restrict__ av,
                            float* __restrict__ al, int N) {
  int i = blockIdx.x * blockDim.x + threadIdx.x;
  if (i >= N * 4) return;
  int n = i >> 2, hh = i & 3;
  const float* row = hm + (size_t)n * 64 + hh * 16;
  const float* a = av + hh * 16;
  float s = 0.f;
#pragma unroll
  for (int c = 0; c < 16; ++c) s += row[c] * a[c];
  al[i] = s;
}

__device__ __forceinline__ float leaky02(float x) { return x > 0.f ? x : 0.2f * x; }

__device__ __forceinline__ void atomicMaxF(float* addr, float val) {
  if (val >= 0.f) atomicMax((int*)addr, __float_as_int(val));
  else            atomicMin((unsigned int*)addr, __float_as_uint(val));
}

// ---------------- edge pass 1: segment max of leaky-relu logits ----------------
__global__ void edge_max_k(const int* __restrict__ es, const int* __restrict__ ed,
                           const float* __restrict__ als, const float* __restrict__ ald,
                           float* __restrict__ amax, int E) {
  int i = blockIdx.x * blockDim.x + threadIdx.x;
  if (i >= E * 4) return;
  int e = i >> 2, hh = i & 3;
  int d = ed[e];
  float a = leaky02(als[(size_t)es[e] * 4 + hh] + ald[(size_t)d * 4 + hh]);
  atomicMaxF(&amax[(size_t)d * 4 + hh], a);
}

__global__ void fix_amax_k(float* __restrict__ amax, int n) {
  int i = blockIdx.x * blockDim.x + threadIdx.x;
  if (i >= n) return;
  float a = amax[i];
  if (!__builtin_isfinite(a)) amax[i] = 0.f;
}

// ---------------- edge pass 2: exp-sum denominators ----------------
__global__ void edge_expsum_k(const int* __restrict__ es, const int* __restrict__ ed,
                              const float* __restrict__ als, const float* __restrict__ ald,
                              const float* __restrict__ amax, float* __restrict__ den,
                              int E) {
  int i = blockIdx.x * blockDim.x + threadIdx.x;
  if (i >= E * 4) return;
  int e = i >> 2, hh = i & 3;
  int d = ed[e];
  float a = leaky02(als[(size_t)es[e] * 4 + hh] + ald[(size_t)d * 4 + hh]);
  float ex = __expf(a - amax[(size_t)d * 4 + hh]);
  atomicAdd(&den[(size_t)d * 4 + hh], ex);
}

// ---------------- edge pass 3: normalized message scatter-add ----------------
__global__ void edge_scatter_k(const int* __restrict__ es, const int* __restrict__ ed,
                               const float* __restrict__ als, const float* __restrict__ ald,
                               const float* __restrict__ amax, const float* __restrict__ den,
                               const float* __restrict__ hs, float* __restrict__ acc,
                               int E) {
  int i = blockIdx.x * blockDim.x + threadIdx.x;
  if (i >= E * 4) return;
  int e = i >> 2, hh = i & 3;
  int s = es[e], d = ed[e];
  float a = leaky02(als[(size_t)s * 4 + hh] + ald[(size_t)d * 4 + hh]);
  float ex = __expf(a - amax[(size_t)d * 4 + hh]);
  float coef = ex / (den[(size_t)d * 4 + hh] + 1e-16f);
  const float* hr = hs + (size_t)s * 64 + hh * 16;
  float* ar = acc + (size_t)d * 64 + hh * 16;
#pragma unroll
  for (int c = 0; c < 16; ++c) atomicAdd(&ar[c], hr[c] * coef);
}

// ---------------- bias-sum + relu + optional residual + LayerNorm ----------------
__global__ void post_ln(float* __restrict__ h, const float* __restrict__ acc,
                        const float* __restrict__ b0, const float* __restrict__ b1,
                        const float* __restrict__ b2, const float* __restrict__ lnw,
                        const float* __restrict__ lnb, int N, int residual) {
  int n = blockIdx.x * blockDim.x + threadIdx.x;
  if (n >= N) return;
  float v[64];
  float mean = 0.f;
  for (int c = 0; c < 64; ++c) {
    float x = acc[(size_t)n * 64 + c] + b0[c];
    if (b1) x += b1[c];
    if (b2) x += b2[c];
    x = fmaxf(x, 0.f);
    if (residual) x += h[(size_t)n * 64 + c];
    v[c] = x;
    mean += x;
  }
  mean *= (1.f / 64.f);
  float var = 0.f;
  for (int c = 0; c < 64; ++c) { float dd = v[c] - mean; var += dd * dd; }
  var *= (1.f / 64.f);
  float rs = rsqrtf(var + 1e-5f);
  for (int c = 0; c < 64; ++c)
    h[(size_t)n * 64 + c] = (v[c] - mean) * rs * lnw[c] + lnb[c];
}

// ==================== driver ====================
extern "C" void kernel_launch(void* const* d_in, const int* in_sizes, int n_in,
                              void* d_out, int out_size, void* d_ws, size_t ws_size,
                              hipStream_t stream) {
  const float* playlist_x   = (const float*)d_in[0];
  const float* track_x      = (const float*)d_in[1];
  const float* artist_x     = (const float*)d_in[2];
  const float* genre_x      = (const float*)d_in[3];
  const float* emb_playlist = (const float*)d_in[4];
  const float* emb_artist   = (const float*)d_in[5];
  const float* pf_W = (const float*)d_in[6];  const float* pf_b = (const float*)d_in[7];
  const float* tf_W = (const float*)d_in[8];  const float* tf_b = (const float*)d_in[9];
  const float* af_W = (const float*)d_in[10]; const float* af_b = (const float*)d_in[11];
  const float* gf_W = (const float*)d_in[12]; const float* gf_b = (const float*)d_in[13];
  const float* ipWp = (const float*)d_in[14]; const float* ipbp = (const float*)d_in[15];
  const float* ipWt = (const float*)d_in[16]; const float* ipbt = (const float*)d_in[17];
  const float* ipWa = (const float*)d_in[18]; const float* ipba = (const float*)d_in[19];
  const float* ipWg = (const float*)d_in[20]; const float* ipbg = (const float*)d_in[21];
  const float* gat_W  = (const float*)d_in[22];
  const float* gat_b  = (const float*)d_in[23];
  const float* gat_as = (const float*)d_in[24];
  const float* gat_ad = (const float*)d_in[25];
  const float* ln_w   = (const float*)d_in[26];
  const float* ln_b   = (const float*)d_in[27];
  const float* opWp = (const float*)d_in[28]; const float* opbp = (const float*)d_in[29];
  const float* opWt = (const float*)d_in[30]; const float* opbt = (const float*)d_in[31];
  const float* opWa = (const float*)d_in[32]; const float* opba = (const float*)d_in[33];
  const float* opWg = (const float*)d_in[34]; const float* opbg = (const float*)d_in[35];
  const int* playlist_idx = (const int*)d_in[36];
  const int* artist_idx   = (const int*)d_in[37];
  const int* ei[8]; int E[8];
  for (int k = 0; k < 8; ++k) { ei[k] = (const int*)d_in[38 + k]; E[k] = in_sizes[38 + k] / 2; }

  float* out = (float*)d_out;
  float* ws  = (float*)d_ws;
  size_t off = 0;
  auto take = [&](size_t n) { float* p = ws + off; off += n; return p; };
  float* h_p   = take((size_t)NPL * 64);
  float* h_t   = take((size_t)NTR * 64);
  float* h_a   = take((size_t)NAR * 64);
  float* h_g   = take((size_t)NGE * 64);
  float* acc_p = take((size_t)NPL * 64);
  float* acc_t = take((size_t)NTR * 64);
  float* acc_a = take((size_t)NAR * 64);
  float* acc_g = take((size_t)NGE * 64);
  float* hs_buf = take((size_t)NTR * 64);   // also encoder staging
  float* hd_buf = take((size_t)NTR * 64);
  float* als  = take((size_t)NTR * 4);
  float* ald  = take((size_t)NTR * 4);
  float* amax = take((size_t)NTR * 4);
  float* den  = take((size_t)NTR * 4);

  const int B = 256;
  auto blk = [](long long n, int b) { return (unsigned)((n + b - 1) / b); };
  auto gemm = [&](const float* X, const float* W, const float* bias, float* Y, int M, int K) {
    int rowTiles = (M + 15) / 16;
    unsigned blocks = (unsigned)((rowTiles + 7) / 8);   // 8 waves/block, 1 stripe/wave
    wmma_gemm_f32<<<blocks, 256, 0, stream>>>(X, W, bias, Y, M, K);
  };

  // ---- input encoders ----
  small_gemm<<<blk((long long)NPL * 64, B), B, 0, stream>>>(playlist_x, pf_W, pf_b, hd_buf, NPL, 8);
  concat_emb_feat<<<blk((long long)NPL * 128, B), B, 0, stream>>>(emb_playlist, playlist_idx, hd_buf, hs_buf, NPL);
  gemm(hs_buf, ipWp, ipbp, h_p, NPL, 128);

  small_gemm<<<blk((long long)NTR * 64, B), B, 0, stream>>>(track_x, tf_W, tf_b, hs_buf, NTR, 7);
  gemm(hs_buf, ipWt, ipbt, h_t, NTR, 64);

  small_gemm<<<blk((long long)NAR * 64, B), B, 0, stream>>>(artist_x, af_W, af_b, hd_buf, NAR, 1);
  concat_emb_feat<<<blk((long long)NAR * 128, B), B, 0, stream>>>(emb_artist, artist_idx, hd_buf, hs_buf, NAR);
  gemm(hs_buf, ipWa, ipba, h_a, NAR, 128);

  gemm(genre_x, gf_W, gf_b, hs_buf, NGE, 500);
  gemm(hs_buf, ipWg, ipbg, h_g, NGE, 64);

  struct Rel { const float* xs; int Ns; const float* xd; int Nd; int e; float* acc; };
  Rel rels[8] = {
    {h_p, NPL, h_t, NTR, 0, acc_t},   // contains        P->T
    {h_t, NTR, h_p, NPL, 1, acc_p},   // in_playlist     T->P
    {h_t, NTR, h_a, NAR, 2, acc_a},   // by              T->A
    {h_a, NAR, h_t, NTR, 3, acc_t},   // created         A->T
    {h_t, NTR, h_g, NGE, 4, acc_g},   // has_genre       T->G
    {h_g, NGE, h_t, NTR, 5, acc_t},   // includes        G->T
    {h_a, NAR, h_g, NGE, 6, acc_g},   // performs        A->G
    {h_g, NGE, h_a, NAR, 7, acc_a},   // performed_by    G->A
  };

  for (int i = 0; i < 2; ++i) {
    fill_f32<<<blk((long long)NPL * 64, B), B, 0, stream>>>(acc_p, 0.f, NPL * 64);
    fill_f32<<<blk((long long)NTR * 64, B), B, 0, stream>>>(acc_t, 0.f, NTR * 64);
    fill_f32<<<blk((long long)NAR * 64, B), B, 0, stream>>>(acc_a, 0.f, NAR * 64);
    fill_f32<<<blk((long long)NGE * 64, B), B, 0, stream>>>(acc_g, 0.f, NGE * 64);

    for (int r = 0; r < 8; ++r) {
      Rel& R = rels[r];
      const float* W  = gat_W  + (size_t)(i * 8 + r) * 64 * 64;
      const float* As = gat_as + (size_t)(i * 8 + r) * 64;
      const float* Ad = gat_ad + (size_t)(i * 8 + r) * 64;
      const int* es = ei[R.e];
      const int* ed = es + E[R.e];
      int Ee = E[R.e];

      gemm(R.xs, W, nullptr, hs_buf, R.Ns, 64);
      gemm(R.xd, W, nullptr, hd_buf, R.Nd, 64);
      attn_logits<<<blk((long long)R.Ns * 4, B), B, 0, stream>>>(hs_buf, As, als, R.Ns);
      attn_logits<<<blk((long long)R.Nd * 4, B), B, 0, stream>>>(hd_buf, Ad, ald, R.Nd);
      fill_f32<<<blk((long long)R.Nd * 4, B), B, 0, stream>>>(amax, -__builtin_inff(), R.Nd * 4);
      fill_f32<<<blk((long long)R.Nd * 4, B), B, 0, stream>>>(den, 0.f, R.Nd * 4);
      edge_max_k<<<blk((long long)Ee * 4, B), B, 0, stream>>>(es, ed, als, ald, amax, Ee);
      fix_amax_k<<<blk((long long)R.Nd * 4, B), B, 0, stream>>>(amax, R.Nd * 4);
      edge_expsum_k<<<blk((long long)Ee * 4, B), B, 0, stream>>>(es, ed, als, ald, amax, den, Ee);
      edge_scatter_k<<<blk((long long)Ee * 4, B), B, 0, stream>>>(es, ed, als, ald, amax, den, hs_buf, R.acc, Ee);
    }

    const float* lw = ln_w + i * 64;
    const float* lb = ln_b + i * 64;
    auto gb = [&](int r) { return gat_b + (size_t)(i * 8 + r) * 64; };
    post_ln<<<blk(NTR, B), B, 0, stream>>>(h_t, acc_t, gb(0), gb(3), gb(5), lw, lb, NTR, i);
    post_ln<<<blk(NPL, B), B, 0, stream>>>(h_p, acc_p, gb(1), nullptr, nullptr, lw, lb, NPL, i);
    post_ln<<<blk(NAR, B), B, 0, stream>>>(h_a, acc_a, gb(2), gb(7), nullptr, lw, lb, NAR, i);
    post_ln<<<blk(NGE, B), B, 0, stream>>>(h_g, acc_g, gb(4), gb(6), nullptr, lw, lb, NGE, i);
  }

  // ---- output projections into concatenated [NP+NT+NA+NG, 64] ----
  gemm(h_p, opWp, opbp, out, NPL, 64);
  gemm(h_t, opWt, opbt, out + (size_t)NPL * 64, NTR, 64);
  gemm(h_a, opWa, opba, out + (size_t)(NPL + NTR) * 64, NAR, 64);
  gemm(h_g, opWg, opbg, out + (size_t)(NPL + NTR + NAR) * 64, NGE, 64);

  (void)n_in; (void)out_size; (void)ws_size;
}